// SampledKWinners_14362370638074
// MI455X (gfx1250) — compile-verified
//
#include <hip/hip_runtime.h>
#include <hip/hip_bf16.h>

// ---------------------------------------------------------------------------
// SampledKWinners for MI455X (gfx1250).
// Memory-bound: 768 MB min traffic -> ~33us at 23.3 TB/s. No matmul structure,
// so the CDNA5 path that matters is async global->LDS DMA + on-chip radix
// select, keeping global traffic at the exact minimum (read x, read g, write out).
// ---------------------------------------------------------------------------

#define ROW_N   4096
#define KSEL    410u
#define TPB     256
#define EPT     (ROW_N / TPB)   // 16 elements per thread
#define INV_T   0.1f            // 1 / TEMPERATURE

#if __has_builtin(__builtin_amdgcn_global_load_async_to_lds_b128) && \
    __has_builtin(__builtin_amdgcn_s_wait_asynccnt)
#define USE_ASYNC_LDS 1
#else
#define USE_ASYNC_LDS 0
#endif

// Native clang vector types (HIP's float4 is a class type; builtins need these).
typedef int   v4i __attribute__((vector_size(16)));
typedef float v4f __attribute__((ext_vector_type(4)));

// Address-space-qualified pointers for the async DMA builtin
// (param types per clang diagnostic: v4i in AS1 "__device__" / AS3 "__shared__").
typedef __attribute__((address_space(1))) v4i gv4i_t;  // global
typedef __attribute__((address_space(3))) v4i lv4i_t;  // LDS

// Monotonic float -> uint mapping (order preserving).
__device__ __forceinline__ unsigned monokey(float f) {
    unsigned u = __float_as_uint(f);
    return (u & 0x80000000u) ? ~u : (u | 0x80000000u);
}

__global__ __launch_bounds__(TPB)
void skw_row_kernel(const float* __restrict__ x,
                    const float* __restrict__ gum,
                    float* __restrict__ out,
                    float* __restrict__ ent_ws)
{
    __shared__ __align__(16) float    s_x[ROW_N];     // 16 KB: x row
    __shared__ __align__(16) unsigned s_key[ROW_N];   // 16 KB: gumbel bits, then keys
    __shared__ unsigned s_hist[TPB];                  // radix histogram / tie scan
    __shared__ float    s_redf[TPB];
    __shared__ float    s_redf2[TPB];
    __shared__ unsigned s_state[2];                   // {prefix, remaining-rank}

    const int    t    = threadIdx.x;
    const size_t base = (size_t)blockIdx.x * ROW_N;

    // ---- Stage 1: stream both rows into LDS via CDNA5 async DMA ----
#if USE_ASYNC_LDS
    #pragma unroll
    for (int j = 0; j < 4; ++j) {
        const int off = (j * TPB + t) * 4;            // float index, 16B granules
        __builtin_amdgcn_global_load_async_to_lds_b128(
            (gv4i_t*)(x   + base + off), (lv4i_t*)(&s_x[off]),   0, 0);
        __builtin_amdgcn_global_load_async_to_lds_b128(
            (gv4i_t*)(gum + base + off), (lv4i_t*)(&s_key[off]), 0, 0);
    }
    __builtin_amdgcn_s_wait_asynccnt(0);
#else
    #pragma unroll
    for (int j = 0; j < 4; ++j) {
        const int off = (j * TPB + t) * 4;
        *(v4f*)(void*)(&s_x[off])   = *(const v4f*)(x   + base + off);
        *(v4f*)(void*)(&s_key[off]) = *(const v4f*)(gum + base + off);
    }
#endif
    __syncthreads();

    // ---- Stage 2: build monotonic keys (logit + gumbel), row max of logits ----
    float lmax = -3.402823466e38f;
    #pragma unroll
    for (int j = 0; j < EPT; ++j) {
        const int i = t * EPT + j;
        const float l = s_x[i] * INV_T;
        const float g = __uint_as_float(s_key[i]);
        s_key[i] = monokey(l + g);
        lmax = fmaxf(lmax, l);
    }
    s_redf[t] = lmax;
    __syncthreads();
    for (int s = TPB / 2; s > 0; s >>= 1) {
        if (t < s) s_redf[t] = fmaxf(s_redf[t], s_redf[t + s]);
        __syncthreads();
    }
    const float rmax = s_redf[0];
    __syncthreads();

    // ---- Stage 3: entropy = log(S) - (sum e*(l-max))/S ----
    float se = 0.0f, sel = 0.0f;
    #pragma unroll
    for (int j = 0; j < EPT; ++j) {
        const int i = t * EPT + j;
        const float l = s_x[i] * INV_T - rmax;
        const float e = __expf(l);
        se  += e;
        sel += e * l;
    }
    s_redf[t] = se;
    s_redf2[t] = sel;
    __syncthreads();
    for (int s = TPB / 2; s > 0; s >>= 1) {
        if (t < s) { s_redf[t] += s_redf[t + s]; s_redf2[t] += s_redf2[t + s]; }
        __syncthreads();
    }
    if (t == 0) {
        const float S = s_redf[0];
        ent_ws[blockIdx.x] = __logf(S) - s_redf2[0] / S;
        s_state[0] = 0u;      // radix prefix
        s_state[1] = KSEL;    // rank (1-based from top) we seek
    }

    // ---- Stage 4: exact K-th-largest key via 4x8-bit LDS radix select ----
    for (int shift = 24; shift >= 0; shift -= 8) {
        s_hist[t] = 0u;
        __syncthreads();
        const unsigned prefix = s_state[0];
        const unsigned hmask  = (shift == 24) ? 0u : (0xFFFFFFFFu << (shift + 8));
        #pragma unroll
        for (int j = 0; j < EPT; ++j) {
            const unsigned k = s_key[t * EPT + j];
            if (((k ^ prefix) & hmask) == 0u)
                atomicAdd(&s_hist[(k >> shift) & 0xFFu], 1u);
        }
        __syncthreads();
        if (t == 0) {
            unsigned rem = s_state[1];
            int b = 255;
            for (; b > 0; --b) {
                const unsigned c = s_hist[b];
                if (c >= rem) break;
                rem -= c;
            }
            s_state[0] = prefix | ((unsigned)b << shift);
            s_state[1] = rem;
        }
        __syncthreads();
    }
    const unsigned kth    = s_state[0];
    const unsigned r_ties = s_state[1];   // how many == kth to keep (lowest index first)

    // ---- Stage 5: index-ordered tie scan (matches jax top_k tie-break) ----
    unsigned eq = 0;
    #pragma unroll
    for (int j = 0; j < EPT; ++j) eq += (s_key[t * EPT + j] == kth) ? 1u : 0u;
    s_hist[t] = eq;
    __syncthreads();
    for (int s = 1; s < TPB; s <<= 1) {   // Hillis-Steele inclusive scan
        const unsigned v = (t >= s) ? s_hist[t - s] : 0u;
        __syncthreads();
        s_hist[t] += v;
        __syncthreads();
    }
    unsigned running = s_hist[t] - eq;    // equals strictly before my chunk

    // ---- Stage 6: masked output, nontemporal b128 stores ----
    #pragma unroll
    for (int j4 = 0; j4 < EPT / 4; ++j4) {
        v4f f4;
        #pragma unroll
        for (int q = 0; q < 4; ++q) {
            const int i = t * EPT + j4 * 4 + q;
            const unsigned k = s_key[i];
            bool selbit;
            if (k > kth)       selbit = true;
            else if (k == kth) { selbit = (running < r_ties); ++running; }
            else               selbit = false;
            f4[q] = selbit ? s_x[i] : 0.0f;
        }
        __builtin_nontemporal_store(f4, (v4f*)(out + base + t * EPT + j4 * 4));
    }
}

// Deterministic mean of per-row entropies -> single scalar.
__global__ __launch_bounds__(256)
void skw_ent_reduce(const float* __restrict__ ws, float* __restrict__ out_scalar,
                    int nrows)
{
    __shared__ float red[256];
    float s = 0.0f;
    for (int i = threadIdx.x; i < nrows; i += 256) s += ws[i];
    red[threadIdx.x] = s;
    __syncthreads();
    for (int st = 128; st > 0; st >>= 1) {
        if (threadIdx.x < st) red[threadIdx.x] += red[threadIdx.x + st];
        __syncthreads();
    }
    if (threadIdx.x == 0) *out_scalar = red[0] / (float)nrows;
}

extern "C" void kernel_launch(void* const* d_in, const int* in_sizes, int n_in,
                              void* d_out, int out_size, void* d_ws, size_t ws_size,
                              hipStream_t stream) {
    const float* x   = (const float*)d_in[0];
    const float* gum = (const float*)d_in[1];
    float*       out = (float*)d_out;
    float*       ws  = (float*)d_ws;     // nrows floats of per-row entropy

    const int nrows = in_sizes[0] / ROW_N;   // 16384

    skw_row_kernel<<<nrows, TPB, 0, stream>>>(x, gum, out, ws);
    skw_ent_reduce<<<1, 256, 0, stream>>>(ws, out + (size_t)nrows * ROW_N, nrows);
}